// RobustSingleTelescopeGNN_12618613915690
// MI455X (gfx1250) — compile-verified
//
#include <hip/hip_runtime.h>
#include <hip/hip_bf16.h>

#define N_NODES 100000
#define N_EDGES 3200000
#define N_GRAPHS 512
#define HID 64
#define BN_EPS 1e-5f

typedef __attribute__((ext_vector_type(2))) float v2f;
typedef __attribute__((ext_vector_type(8))) float v8f;
typedef __attribute__((ext_vector_type(4))) unsigned int u32x4;
typedef __attribute__((ext_vector_type(8))) int i32x8;
typedef __attribute__((ext_vector_type(4))) int i32x4;

// ---------------------------------------------------------------- utilities

__global__ void zero_f32_kernel(float* __restrict__ p, int n) {
    int i = blockIdx.x * blockDim.x + threadIdx.x;
    if (i < n) p[i] = 0.0f;
}

__global__ void edge_deg_kernel(const int* __restrict__ dst, float* __restrict__ deg, int e) {
    int i = blockIdx.x * blockDim.x + threadIdx.x;
    if (i < e) atomicAdd(&deg[dst[i]], 1.0f);
}

__global__ void make_dinv_kernel(float* __restrict__ d, int n) {
    int i = blockIdx.x * blockDim.x + threadIdx.x;
    if (i < n) d[i] = rsqrtf(d[i] + 1.0f);   // +1 self loop
}

// layer1: h = x (N x 1) @ w1 (1 x 64)  -> outer product
__global__ void feat1_kernel(const float* __restrict__ x, const float* __restrict__ w1,
                             float* __restrict__ h, int n) {
    int i = blockIdx.x * blockDim.x + threadIdx.x;
    if (i < n * HID) {
        int node = i >> 6, j = i & 63;
        h[i] = x[node] * w1[j];
    }
}

// --------------------------------------------------------- fp32 WMMA GEMM
// D[M x 64] = A[M x 64] @ W[64 x 64], exact fp32 via V_WMMA_F32_16X16X4_F32.
// 256 threads = 8 waves; each wave computes a 16-row x 64-col slab.
// W staged in LDS with stride 72: row k and k+2 (read by lane halves of one
// ds_load) map to disjoint 16-bank sets (2*72 mod 64 == 16) -> conflict-free.
// Staging is done by the Tensor Data Mover: one TENSOR_LOAD_TO_LDS from wave 0
// with pad_interval=64 DWORDs / pad_amount=8 DWORDs reproduces the 72-DWORD
// padded row layout in hardware; completion via s_wait_tensorcnt + barrier.
__global__ __launch_bounds__(256) void gemm_n64_wmma_kernel(
        const float* __restrict__ A, const float* __restrict__ W,
        float* __restrict__ D, int M) {
    __shared__ float sW[64 * 72];
    const int t = threadIdx.x;

#if __has_builtin(__builtin_amdgcn_tensor_load_to_lds) && __has_builtin(__builtin_amdgcn_s_wait_tensorcnt)
    if (t < 32) {                                   // wave 0 issues the DMA
        unsigned long long ga = (unsigned long long)(uintptr_t)W;
        unsigned int ldsa = (unsigned int)(uintptr_t)sW;  // flat LDS addr[31:0] == LDS offset
        u32x4 g0;
        g0[0] = 1u;                                 // count=1, user mode
        g0[1] = ldsa;                               // lds_addr
        g0[2] = (unsigned int)ga;                   // global_addr[31:0]
        g0[3] = (unsigned int)((ga >> 32) & 0x1FFFFFFu) | (2u << 30);  // ga[56:32] | type=2
        i32x8 g1;
        g1[0] = (2 << 16)      // data_size = 4B
              | (1 << 20)      // pad_enable
              | (5 << 22)      // pad_interval: every 64 DWORDs
              | (7 << 25);     // pad_amount: 8 DWORDs  -> 72-DWORD LDS rows
        g1[1] = (HID & 0xFFFF) << 16;               // tensor_dim0 lo16
        g1[2] = (HID << 16);                        // tensor_dim0 hi16=0 | tensor_dim1 lo16
        g1[3] = (HID << 16);                        // tensor_dim1 hi16=0 | tile_dim0=64
        g1[4] = HID;                                // tile_dim1=64 | tile_dim2=0
        g1[5] = HID;                                // tensor_dim0_stride lo32 = 64 elements
        g1[6] = 0;
        g1[7] = 0;
        i32x4 gz  = {0, 0, 0, 0};                   // 2-D tile: groups 2/3 unused
        i32x8 gz8 = {0, 0, 0, 0, 0, 0, 0, 0};
        __builtin_amdgcn_tensor_load_to_lds(g0, g1, gz, gz, gz8, 0);
        __builtin_amdgcn_s_wait_tensorcnt(0);
    }
    __syncthreads();
#else
    for (int i = t; i < 64 * 64; i += 256)
        sW[(i >> 6) * 72 + (i & 63)] = W[i];
    __syncthreads();
#endif

    const int wave  = t >> 5;
    const int lane  = t & 31;
    const int lhalf = lane >> 4;   // 0: lanes 0-15 (K+0,K+1) | 1: lanes 16-31 (K+2,K+3)
    const int l16   = lane & 15;

    const int m0   = blockIdx.x * 128 + wave * 16;
    int arow = m0 + l16;
    if (arow > M - 1) arow = M - 1;            // clamp loads; EXEC stays all-1 for WMMA
    const float* arowp = A + (size_t)arow * HID + lhalf * 2;

    v8f acc0 = {}, acc1 = {}, acc2 = {}, acc3 = {};
    for (int k = 0; k < 64; k += 4) {
        const float2 av = *(const float2*)(arowp + k);   // A frag: 16x4 layout
        v2f a; a.x = av.x; a.y = av.y;
        const int r0 = k + lhalf * 2;
        {   // cols 0..15
            v2f b; b.x = sW[r0 * 72 + l16];      b.y = sW[(r0 + 1) * 72 + l16];
            acc0 = __builtin_amdgcn_wmma_f32_16x16x4_f32(false, a, false, b, (short)0, acc0, false, false);
        }
        {   // cols 16..31
            v2f b; b.x = sW[r0 * 72 + 16 + l16]; b.y = sW[(r0 + 1) * 72 + 16 + l16];
            acc1 = __builtin_amdgcn_wmma_f32_16x16x4_f32(false, a, false, b, (short)0, acc1, false, false);
        }
        {   // cols 32..47
            v2f b; b.x = sW[r0 * 72 + 32 + l16]; b.y = sW[(r0 + 1) * 72 + 32 + l16];
            acc2 = __builtin_amdgcn_wmma_f32_16x16x4_f32(false, a, false, b, (short)0, acc2, false, false);
        }
        {   // cols 48..63
            v2f b; b.x = sW[r0 * 72 + 48 + l16]; b.y = sW[(r0 + 1) * 72 + 48 + l16];
            acc3 = __builtin_amdgcn_wmma_f32_16x16x4_f32(false, a, false, b, (short)0, acc3, false, false);
        }
    }

    // C/D layout: VGPR r -> row m0 + r + 8*lhalf, col = tile*16 + l16
    const int rbase = m0 + lhalf * 8;
    for (int r = 0; r < 8; ++r) {
        int row = rbase + r;
        if (row < M) {
            float* dp = D + (size_t)row * HID + l16;
            dp[0]  = acc0[r];
            dp[16] = acc1[r];
            dp[32] = acc2[r];
            dp[48] = acc3[r];
        }
    }
}

// ------------------------------------------------------------- aggregation
// One wave per edge; each lane owns 2 consecutive features (32*2 = 64).
__global__ __launch_bounds__(256) void aggregate_kernel(
        const float* __restrict__ h, const int* __restrict__ src,
        const int* __restrict__ dst, const float* __restrict__ dinv,
        float* __restrict__ agg, int e) {
    long long gid = (long long)blockIdx.x * blockDim.x + threadIdx.x;
    int edge = (int)(gid >> 5);
    int lane = (int)(gid & 31);
    if (edge >= e) return;
    // prefetch upcoming edge indices (speculative: OOB silently dropped)
    __builtin_prefetch(&src[edge + 8192], 0, 0);
    __builtin_prefetch(&dst[edge + 8192], 0, 0);
    int s = src[edge], d = dst[edge];
    float c = dinv[s] * dinv[d];
    const float2 v = *(const float2*)(h + (size_t)s * HID + lane * 2);
    float* ap = agg + (size_t)d * HID + lane * 2;
    atomicAdd(ap + 0, v.x * c);
    atomicAdd(ap + 1, v.y * c);
}

// fused: + self-loop message + bias, BatchNorm(running stats), ReLU
__global__ void combine_kernel(const float* __restrict__ agg, const float* __restrict__ h,
                               const float* __restrict__ dinv, const float* __restrict__ bias,
                               const float* __restrict__ g, const float* __restrict__ bb,
                               const float* __restrict__ rm, const float* __restrict__ rv,
                               float* __restrict__ out, int n) {
    int i = blockIdx.x * blockDim.x + threadIdx.x;
    if (i >= n * HID) return;
    int node = i >> 6, j = i & 63;
    float di = dinv[node];
    float v = agg[i] + h[i] * (di * di) + bias[j];
    v = (v - rm[j]) * (g[j] * rsqrtf(rv[j] + BN_EPS)) + bb[j];
    out[i] = fmaxf(v, 0.0f);
}

// ------------------------------------------------------------------ pooling

__global__ void pool_cnt_kernel(const int* __restrict__ batch, float* __restrict__ cnt, int n) {
    int i = blockIdx.x * blockDim.x + threadIdx.x;
    if (i < n) atomicAdd(&cnt[batch[i]], 1.0f);
}

// wave per node: mean-sum via atomicAdd, max via int atomicMax (values >= 0 post-ReLU)
__global__ __launch_bounds__(256) void pool_accum_kernel(
        const float* __restrict__ cur, const int* __restrict__ batch,
        float* __restrict__ zsum, int* __restrict__ zmax, int n) {
    long long gid = (long long)blockIdx.x * blockDim.x + threadIdx.x;
    int node = (int)(gid >> 5);
    int lane = (int)(gid & 31);
    if (node >= n) return;
    int gph = batch[node];
    const float2 v = *(const float2*)(cur + (size_t)node * HID + lane * 2);
    atomicAdd(&zsum[gph * HID + lane * 2 + 0], v.x);
    atomicAdd(&zsum[gph * HID + lane * 2 + 1], v.y);
    atomicMax(&zmax[gph * HID + lane * 2 + 0], __float_as_int(v.x));
    atomicMax(&zmax[gph * HID + lane * 2 + 1], __float_as_int(v.y));
}

__global__ void finalize_z_kernel(const float* __restrict__ zsum, const int* __restrict__ zmax,
                                  const float* __restrict__ cnt, float* __restrict__ z) {
    int i = blockIdx.x * blockDim.x + threadIdx.x;   // over G*64
    if (i >= N_GRAPHS * HID) return;
    int gph = i >> 6, j = i & 63;
    float c = fmaxf(cnt[gph], 1.0f);
    z[gph * 2 * HID + j]        = zsum[i] / c;
    z[gph * 2 * HID + HID + j]  = __int_as_float(zmax[i]);
}

// --------------------------------------------------------------- MLP head

__global__ void head1_kernel(const float* __restrict__ z, const float* __restrict__ wl1,
                             const float* __restrict__ bl1,
                             const float* __restrict__ g, const float* __restrict__ bb,
                             const float* __restrict__ rm, const float* __restrict__ rv,
                             float* __restrict__ hd) {
    int i = blockIdx.x * blockDim.x + threadIdx.x;   // over G*64
    if (i >= N_GRAPHS * HID) return;
    int gph = i >> 6, j = i & 63;
    const float* zp = z + gph * 2 * HID;
    float acc = bl1[j];
    #pragma unroll 8
    for (int k = 0; k < 2 * HID; ++k)
        acc += zp[k] * wl1[k * HID + j];
    acc = (acc - rm[j]) * (g[j] * rsqrtf(rv[j] + BN_EPS)) + bb[j];
    hd[i] = fmaxf(acc, 0.0f);
}

__global__ void head2_kernel(const float* __restrict__ hd, const float* __restrict__ wl2,
                             const float* __restrict__ bl2, float* __restrict__ out) {
    int i = blockIdx.x * blockDim.x + threadIdx.x;   // over G*2
    if (i >= N_GRAPHS * 2) return;
    int gph = i >> 1, j = i & 1;
    const float* hp = hd + gph * HID;
    float acc = bl2[j];
    #pragma unroll 8
    for (int k = 0; k < HID; ++k)
        acc += hp[k] * wl2[k * 2 + j];
    out[i] = acc;
}

// ------------------------------------------------------------------ driver

static inline int cdiv(long long a, int b) { return (int)((a + b - 1) / b); }

extern "C" void kernel_launch(void* const* d_in, const int* in_sizes, int n_in,
                              void* d_out, int out_size, void* d_ws, size_t ws_size,
                              hipStream_t stream) {
    const float* x    = (const float*)d_in[0];
    const int*   src  = (const int*)d_in[1];
    const int*   dst  = (const int*)d_in[2];
    const int*   batch= (const int*)d_in[3];
    const float* w1   = (const float*)d_in[4];
    const float* b1   = (const float*)d_in[5];
    const float* w2   = (const float*)d_in[6];
    const float* b2   = (const float*)d_in[7];
    const float* w3   = (const float*)d_in[8];
    const float* b3   = (const float*)d_in[9];
    const float* wl1  = (const float*)d_in[10];
    const float* bl1  = (const float*)d_in[11];
    const float* wl2  = (const float*)d_in[12];
    const float* bl2  = (const float*)d_in[13];
    const float* bn_g[4]  = {(const float*)d_in[14], (const float*)d_in[18], (const float*)d_in[22], (const float*)d_in[26]};
    const float* bn_b[4]  = {(const float*)d_in[15], (const float*)d_in[19], (const float*)d_in[23], (const float*)d_in[27]};
    const float* bn_rm[4] = {(const float*)d_in[16], (const float*)d_in[20], (const float*)d_in[24], (const float*)d_in[28]};
    const float* bn_rv[4] = {(const float*)d_in[17], (const float*)d_in[21], (const float*)d_in[25], (const float*)d_in[29]};
    float* out = (float*)d_out;

    const int N = N_NODES, E = N_EDGES, G = N_GRAPHS;

    // workspace layout (floats)
    float* ws   = (float*)d_ws;
    float* dinv = ws;                               // N
    float* h    = dinv + N;                         // N*64
    float* agg  = h    + (size_t)N * HID;           // N*64
    float* cur  = agg  + (size_t)N * HID;           // N*64
    float* zsum = cur  + (size_t)N * HID;           // G*64
    int*   zmax = (int*)(zsum + (size_t)G * HID);   // G*64  (bit-pattern max, vals >= 0)
    float* cnt  = (float*)(zmax + (size_t)G * HID); // G
    float* z    = cnt + G;                          // G*128
    float* hd   = z   + (size_t)G * 2 * HID;        // G*64

    const int B = 256;

    // degrees -> dinv
    zero_f32_kernel<<<cdiv(N, B), B, 0, stream>>>(dinv, N);
    edge_deg_kernel<<<cdiv(E, B), B, 0, stream>>>(dst, dinv, E);
    make_dinv_kernel<<<cdiv(N, B), B, 0, stream>>>(dinv, N);

    // ---- layer 1 (1 -> 64)
    feat1_kernel<<<cdiv((long long)N * HID, B), B, 0, stream>>>(x, w1, h, N);
    zero_f32_kernel<<<cdiv((long long)N * HID, B), B, 0, stream>>>(agg, N * HID);
    aggregate_kernel<<<cdiv((long long)E * 32, B), B, 0, stream>>>(h, src, dst, dinv, agg, E);
    combine_kernel<<<cdiv((long long)N * HID, B), B, 0, stream>>>(
        agg, h, dinv, b1, bn_g[0], bn_b[0], bn_rm[0], bn_rv[0], cur, N);

    // ---- layer 2 (64 -> 64, fp32 WMMA, TDM-staged weights)
    gemm_n64_wmma_kernel<<<cdiv(N, 128), 256, 0, stream>>>(cur, w2, h, N);
    zero_f32_kernel<<<cdiv((long long)N * HID, B), B, 0, stream>>>(agg, N * HID);
    aggregate_kernel<<<cdiv((long long)E * 32, B), B, 0, stream>>>(h, src, dst, dinv, agg, E);
    combine_kernel<<<cdiv((long long)N * HID, B), B, 0, stream>>>(
        agg, h, dinv, b2, bn_g[1], bn_b[1], bn_rm[1], bn_rv[1], cur, N);

    // ---- layer 3 (64 -> 64, fp32 WMMA, TDM-staged weights)
    gemm_n64_wmma_kernel<<<cdiv(N, 128), 256, 0, stream>>>(cur, w3, h, N);
    zero_f32_kernel<<<cdiv((long long)N * HID, B), B, 0, stream>>>(agg, N * HID);
    aggregate_kernel<<<cdiv((long long)E * 32, B), B, 0, stream>>>(h, src, dst, dinv, agg, E);
    combine_kernel<<<cdiv((long long)N * HID, B), B, 0, stream>>>(
        agg, h, dinv, b3, bn_g[2], bn_b[2], bn_rm[2], bn_rv[2], cur, N);

    // ---- pooling (zsum | zmax | cnt are contiguous -> one zero fill)
    zero_f32_kernel<<<cdiv((long long)G * (2 * HID + 1), B), B, 0, stream>>>(zsum, G * (2 * HID + 1));
    pool_cnt_kernel<<<cdiv(N, B), B, 0, stream>>>(batch, cnt, N);
    pool_accum_kernel<<<cdiv((long long)N * 32, B), B, 0, stream>>>(cur, batch, zsum, zmax, N);
    finalize_z_kernel<<<cdiv((long long)G * HID, B), B, 0, stream>>>(zsum, zmax, cnt, z);

    // ---- MLP head
    head1_kernel<<<cdiv((long long)G * HID, B), B, 0, stream>>>(
        z, wl1, bl1, bn_g[3], bn_b[3], bn_rm[3], bn_rv[3], hd);
    head2_kernel<<<cdiv((long long)G * 2, B), B, 0, stream>>>(hd, wl2, bl2, out);
}